// GNNClassifier_47356309406183
// MI455X (gfx1250) — compile-verified
//
#include <hip/hip_runtime.h>
#include <hip/hip_bf16.h>

typedef float v2f __attribute__((ext_vector_type(2)));
typedef float v8f __attribute__((ext_vector_type(8)));

#define HDIM 64
#define CDIM 10
#define NEG_SLOPE 0.01f
#define TEMP 0.05f
#define EPSV 1e-16f

// ---------------- utility ----------------
__global__ void kzero(float* __restrict__ p, size_t n) {
  size_t i = (size_t)blockIdx.x * blockDim.x + threadIdx.x;
  size_t stride = (size_t)gridDim.x * blockDim.x;
  for (; i < n; i += stride) p[i] = 0.0f;
}

__device__ __forceinline__ void atomAddF(float* p, float v) {
  unsafeAtomicAdd(p, v);   // lowers to global_atomic_add_f32
}

// ---------------- GCN normalization ----------------
__global__ void kdeg(const int* __restrict__ col, const float* __restrict__ ew,
                     float* __restrict__ deg, int E) {
  int e = blockIdx.x * blockDim.x + threadIdx.x;
  if (e < E) atomAddF(&deg[col[e]], ew[e]);
}

__global__ void kdis(const float* __restrict__ deg, float* __restrict__ dis, int N) {
  int i = blockIdx.x * blockDim.x + threadIdx.x;
  if (i < N) dis[i] = rsqrtf(deg[i] + 1.0f);   // +1 = self loop weight; always > 0
}

__global__ void knorm(const int* __restrict__ row, const int* __restrict__ col,
                      const float* __restrict__ ew, const float* __restrict__ dis,
                      float* __restrict__ nrm, int E) {
  int e = blockIdx.x * blockDim.x + threadIdx.x;
  if (e < E) nrm[e] = dis[row[e]] * ew[e] * dis[col[e]];
}

// ---------------- fp32 WMMA GEMM: C[N x 64] = A[N x 64] * W[64 x 64] ----------------
// block = 256 threads = 8 waves; block tile = 32 rows x 64 cols (2x4 wave tiles of 16x16)
__global__ __launch_bounds__(256) void kgemm(const float* __restrict__ A,
                                             const float* __restrict__ W,
                                             float* __restrict__ C, int nrows) {
  __shared__ float sW[64 * 64];   // 16 KB
  __shared__ float sA[32 * 64];   // 8 KB
  const int tid = threadIdx.x;
  const int rowbase = blockIdx.x * 32;

  // stage W (4096 floats) as float4
  for (int i = tid; i < 64 * 64 / 4; i += 256)
    ((float4*)sW)[i] = ((const float4*)W)[i];
  // stage 32 rows of A
  for (int i = tid; i < 32 * 64 / 4; i += 256) {
    int r = i >> 4;               // 16 float4 per row
    int c4 = i & 15;
    int row = rowbase + r;
    if (row >= nrows) row = nrows - 1;    // clamp (N divisible by 32 anyway)
    ((float4*)sA)[i] = ((const float4*)(A + (size_t)row * 64))[c4];
  }
  __syncthreads();

  const int wave = tid >> 5;
  const int lane = tid & 31;
  const int rowTile = (wave >> 2) * 16;   // 0 or 16
  const int colTile = (wave & 3) * 16;    // 0,16,32,48
  const int l15 = lane & 15;
  const int khalf = (lane >> 4) * 2;      // lanes 0-15 -> K{0,1}; lanes 16-31 -> K{2,3}

  v8f acc = {};
#pragma unroll
  for (int k = 0; k < 64; k += 4) {
    v2f a, b;
    const float* ap = &sA[(rowTile + l15) * 64 + k + khalf];
    a.x = ap[0];
    a.y = ap[1];
    b.x = sW[(k + khalf) * 64 + colTile + l15];
    b.y = sW[(k + khalf + 1) * 64 + colTile + l15];
    acc = __builtin_amdgcn_wmma_f32_16x16x4_f32(false, a, false, b,
                                                (short)0, acc, false, false);
  }

  // C/D layout: VGPR r: lanes 0-15 -> M=r, lanes 16-31 -> M=r+8; N = lane&15
  const int colOut = colTile + l15;
  const int rowAdd = (lane >> 4) * 8;
#pragma unroll
  for (int r = 0; r < 8; ++r) {
    int row = rowbase + rowTile + rowAdd + r;
    if (row < nrows) C[(size_t)row * 64 + colOut] = acc[r];
  }
}

// ---------------- edge aggregation: hagg[c] += w * hmm[r]  (incl. self loops) -------
// one wave per item; lane handles 2 features (H=64)
__global__ void kagg(const int* __restrict__ row, const int* __restrict__ col,
                     const float* __restrict__ nrm, const float* __restrict__ dis,
                     const float* __restrict__ hmm, float* __restrict__ hagg,
                     int E, int N) {
  long long gid = (long long)blockIdx.x * blockDim.x + threadIdx.x;
  int item = (int)(gid >> 5);
  int lane = (int)(gid & 31);
  if (item >= E + N) return;
  int r, c;
  float w;
  if (item < E) {
    r = row[item];
    c = col[item];
    w = nrm[item];
  } else {
    r = c = item - E;
    float d = dis[r];
    w = d * d;               // self-loop norm = 1/deg
  }
  const float2 v = *(const float2*)(hmm + (size_t)r * 64 + lane * 2);
  float* dst = hagg + (size_t)c * 64 + lane * 2;
  atomAddF(dst, w * v.x);
  atomAddF(dst + 1, w * v.y);
}

// ---------------- bias + optional leaky relu ----------------
__global__ void kbias(float* __restrict__ h, const float* __restrict__ bias,
                      size_t n, int act) {
  size_t i = (size_t)blockIdx.x * blockDim.x + threadIdx.x;
  if (i >= n) return;
  float v = h[i] + bias[i & 63];
  if (act) v = (v > 0.0f) ? v : NEG_SLOPE * v;
  h[i] = v;
}

// ---------------- smooth-max node weight ----------------
__global__ void kmax(const int* __restrict__ row, const int* __restrict__ col,
                     const float* __restrict__ ew, unsigned int* __restrict__ m,
                     int E) {
  int i = blockIdx.x * blockDim.x + threadIdx.x;
  if (i >= 2 * E) return;
  int idx = (i < E) ? row[i] : col[i - E];
  float w = ew[(i < E) ? i : (i - E)];
  // weights are in [0,1): uint compare == float compare; init value 0 matches
  // the reference's where(isfinite(m), m, 0) for isolated nodes.
  atomicMax(&m[idx], __float_as_uint(w));
}

__global__ void kexp(const int* __restrict__ row, const int* __restrict__ col,
                     const float* __restrict__ ew, const float* __restrict__ m,
                     float* __restrict__ den, float* __restrict__ num, int E) {
  int i = blockIdx.x * blockDim.x + threadIdx.x;
  if (i >= 2 * E) return;
  int idx = (i < E) ? row[i] : col[i - E];
  float w = ew[(i < E) ? i : (i - E)];
  float a = __expf((w - m[idx]) / TEMP);
  atomAddF(&den[idx], a);
  atomAddF(&num[idx], w * a);
}

__global__ void knw(const float* __restrict__ num, const float* __restrict__ den,
                    float* __restrict__ nw, int N) {
  int i = blockIdx.x * blockDim.x + threadIdx.x;
  if (i < N) {
    float d = den[i];
    nw[i] = (d > 0.0f) ? num[i] / fmaxf(d, EPSV) : 0.0f;
  }
}

// ---------------- weighted pooling ----------------
__global__ void kpool(const float* __restrict__ h3, const float* __restrict__ nw,
                      const int* __restrict__ batch, float* __restrict__ s,
                      float* __restrict__ wsum, int N) {
  long long gid = (long long)blockIdx.x * blockDim.x + threadIdx.x;
  int node = (int)(gid >> 5);
  int lane = (int)(gid & 31);
  if (node >= N) return;
  int b = batch[node];
  float w = nw[node];
  const float2 v = *(const float2*)(h3 + (size_t)node * 64 + lane * 2);
  float* dst = s + (size_t)b * 64 + lane * 2;
  atomAddF(dst, w * v.x);
  atomAddF(dst + 1, w * v.y);
  if (lane == 0) atomAddF(&wsum[b], w);
}

// ---------------- classifier head (one block per graph) ----------------
__global__ __launch_bounds__(128) void khead(const float* __restrict__ s,
                                             const float* __restrict__ wsum,
                                             const float* __restrict__ Wlin,
                                             const float* __restrict__ blin,
                                             const float* __restrict__ Wout,
                                             const float* __restrict__ bout,
                                             float* __restrict__ out, int B) {
  int b = blockIdx.x;
  int t = threadIdx.x;   // 128 threads
  __shared__ float feats[128];
  __shared__ float hsh[64];
  __shared__ float lg[CDIM];

  const size_t LOG_OFF = 0;
  const size_t PRB_OFF = (size_t)B * CDIM;
  const size_t FEA_OFF = (size_t)2 * B * CDIM;
  const size_t EMB_OFF = FEA_OFF + (size_t)B * 2 * HDIM;
  const size_t HOU_OFF = EMB_OFF + (size_t)B * HDIM;

  float inv = 1.0f / fmaxf(wsum[b], EPSV);
  float f = (t < 64) ? s[(size_t)b * 64 + t] : s[(size_t)b * 64 + (t - 64)] * inv;
  feats[t] = f;
  out[FEA_OFF + (size_t)b * 128 + t] = f;
  __syncthreads();

  if (t < 64) {
    float acc = blin[t];
#pragma unroll 4
    for (int k = 0; k < 128; ++k) acc += feats[k] * Wlin[k * 64 + t];
    out[EMB_OFF + (size_t)b * 64 + t] = acc;
    float h = fmaxf(acc, 0.0f);
    hsh[t] = h;
    out[HOU_OFF + (size_t)b * 64 + t] = h;
  }
  __syncthreads();

  if (t < CDIM) {
    float acc = bout[t];
#pragma unroll
    for (int k = 0; k < 64; ++k) acc += hsh[k] * Wout[k * CDIM + t];
    lg[t] = acc;
    out[LOG_OFF + (size_t)b * CDIM + t] = acc;
  }
  __syncthreads();

  if (t == 0) {
    float mx = lg[0];
    for (int i = 1; i < CDIM; ++i) mx = fmaxf(mx, lg[i]);
    float e[CDIM], sum = 0.0f;
    for (int i = 0; i < CDIM; ++i) { e[i] = __expf(lg[i] - mx); sum += e[i]; }
    float r = 1.0f / sum;
    for (int i = 0; i < CDIM; ++i) out[PRB_OFF + (size_t)b * CDIM + i] = e[i] * r;
  }
}

// ---------------- driver ----------------
extern "C" void kernel_launch(void* const* d_in, const int* in_sizes, int n_in,
                              void* d_out, int out_size, void* d_ws, size_t ws_size,
                              hipStream_t stream) {
  const float* x    = (const float*)d_in[0];
  const int*   ei   = (const int*)d_in[1];
  const float* ew   = (const float*)d_in[2];
  const int*   batch= (const int*)d_in[3];
  const float* W1 = (const float*)d_in[4];  const float* b1 = (const float*)d_in[5];
  const float* W2 = (const float*)d_in[6];  const float* b2 = (const float*)d_in[7];
  const float* W3 = (const float*)d_in[8];  const float* b3 = (const float*)d_in[9];
  const float* Wlin = (const float*)d_in[10]; const float* blin = (const float*)d_in[11];
  const float* Wout = (const float*)d_in[12]; const float* bout = (const float*)d_in[13];
  float* out = (float*)d_out;

  const int N = in_sizes[0] / HDIM;
  const int E = in_sizes[2];
  const int B = out_size / (2 * CDIM + 4 * HDIM);   // 276 floats per graph
  const int* row = ei;
  const int* col = ei + E;

  // workspace layout (floats)
  float* ws   = (float*)d_ws;
  float* dis  = ws;                       // N
  float* nrm  = dis + N;                  // E
  float* buf0 = nrm + E;                  // N*64  (GEMM scratch)
  float* buf1 = buf0 + (size_t)N * 64;    // N*64
  float* buf2 = buf1 + (size_t)N * 64;    // N*64
  float* den  = buf2 + (size_t)N * 64;    // N (also deg scratch)
  float* num  = den + N;                  // N
  float* nw   = num + N;                  // N
  float* mbuf = nw + N;                   // N (uint max)
  float* sbuf = mbuf + N;                 // B*64
  float* wsum = sbuf + (size_t)B * 64;    // B

  const int T = 256;
  auto blocks = [](long long n, int t) { return (int)((n + t - 1) / t); };

  // ---- GCN normalization ----
  kzero<<<1024, T, 0, stream>>>(den, (size_t)N);
  kdeg<<<blocks(E, T), T, 0, stream>>>(col, ew, den, E);
  kdis<<<blocks(N, T), T, 0, stream>>>(den, dis, N);
  knorm<<<blocks(E, T), T, 0, stream>>>(row, col, ew, dis, nrm, E);

  const long long aggThreads = (long long)(E + N) * 32;
  const int gemmBlocks = blocks(N, 32);

  // ---- layer 1: x -> buf1 ----
  kgemm<<<gemmBlocks, 256, 0, stream>>>(x, W1, buf0, N);
  kzero<<<2048, T, 0, stream>>>(buf1, (size_t)N * 64);
  kagg<<<blocks(aggThreads, T), T, 0, stream>>>(row, col, nrm, dis, buf0, buf1, E, N);
  kbias<<<blocks((long long)N * 64, T), T, 0, stream>>>(buf1, b1, (size_t)N * 64, 1);

  // ---- layer 2: buf1 -> buf2 ----
  kgemm<<<gemmBlocks, 256, 0, stream>>>(buf1, W2, buf0, N);
  kzero<<<2048, T, 0, stream>>>(buf2, (size_t)N * 64);
  kagg<<<blocks(aggThreads, T), T, 0, stream>>>(row, col, nrm, dis, buf0, buf2, E, N);
  kbias<<<blocks((long long)N * 64, T), T, 0, stream>>>(buf2, b2, (size_t)N * 64, 1);

  // ---- layer 3: buf2 -> buf1 (no activation) ----
  kgemm<<<gemmBlocks, 256, 0, stream>>>(buf2, W3, buf0, N);
  kzero<<<2048, T, 0, stream>>>(buf1, (size_t)N * 64);
  kagg<<<blocks(aggThreads, T), T, 0, stream>>>(row, col, nrm, dis, buf0, buf1, E, N);
  kbias<<<blocks((long long)N * 64, T), T, 0, stream>>>(buf1, b3, (size_t)N * 64, 0);

  // ---- smooth-max node weights ----
  kzero<<<1024, T, 0, stream>>>(mbuf, (size_t)N);
  kzero<<<1024, T, 0, stream>>>(den, (size_t)N);
  kzero<<<1024, T, 0, stream>>>(num, (size_t)N);
  kmax<<<blocks(2LL * E, T), T, 0, stream>>>(row, col, ew, (unsigned int*)mbuf, E);
  kexp<<<blocks(2LL * E, T), T, 0, stream>>>(row, col, ew, mbuf, den, num, E);
  knw<<<blocks(N, T), T, 0, stream>>>(num, den, nw, N);

  // ---- pooling ----
  kzero<<<64, T, 0, stream>>>(sbuf, (size_t)B * 64);
  kzero<<<1, T, 0, stream>>>(wsum, (size_t)B);
  kpool<<<blocks((long long)N * 32, T), T, 0, stream>>>(buf1, nw, batch, sbuf, wsum, N);

  // ---- head ----
  khead<<<B, 128, 0, stream>>>(sbuf, wsum, Wlin, blin, Wout, bout, out, B);
}